// DeepGCN_12395275616334
// MI455X (gfx1250) — compile-verified
//
#include <hip/hip_runtime.h>

typedef __attribute__((ext_vector_type(2))) float v2f;
typedef __attribute__((ext_vector_type(8))) float v8f;

#define N_CH 128   // K dimension / hidden width (fixed by the model)

// ---------------- degree / normalization ----------------

__global__ void zero_f32(float* __restrict__ p, int n) {
  int i = blockIdx.x * blockDim.x + threadIdx.x;
  if (i < n) p[i] = 0.0f;
}

__global__ void deg_acc(const int* __restrict__ dst, const float* __restrict__ ew,
                        float* __restrict__ deg, int E) {
  int i = blockIdx.x * blockDim.x + threadIdx.x;
  int stride = gridDim.x * blockDim.x;
  for (int e = i; e < E; e += stride)
    unsafeAtomicAdd(&deg[dst[e]], ew[e]);   // global_atomic_add_f32
}

__global__ void make_dinv(const float* __restrict__ deg, float* __restrict__ dinv, int n) {
  int i = blockIdx.x * blockDim.x + threadIdx.x;
  if (i < n) dinv[i] = rsqrtf(deg[i] + 1.0f);   // deg+1 > 0 always
}

// ---------------- WMMA GEMM: C[M x Nc] = A[M x 128] @ W[128 x Nc] (+ bias) ----------------
// grid.x = M/16 (M is a multiple of 16). 256 threads = 8 waves; each wave owns one
// 16x16 output tile (wave w -> columns [16w, 16w+16)).

__global__ __launch_bounds__(256) void gemm_wmma_f32(
    const float* __restrict__ A, const float* __restrict__ W,
    const float* __restrict__ bias, float* __restrict__ C, int Nc) {
  constexpr int LDA = 132;                 // pad: 132 % 64 banks = 4 -> conflict-free column reads
  __shared__ float As[16 * LDA];
  const int row0 = blockIdx.x * 16;
  const int tid = threadIdx.x;

  // cooperative load of the 16x128 A tile (512 float4's, 256 threads)
  {
    const float4* src4 = (const float4*)(A + (size_t)row0 * N_CH);
    for (int i = tid; i < 16 * (N_CH / 4); i += 256) {
      int r = i >> 5, c4 = i & 31;
      *(float4*)(&As[r * LDA + c4 * 4]) = src4[i];   // 528B row pitch keeps 16B alignment
    }
  }
  __syncthreads();

  const int wave = tid >> 5;
  const int lane = tid & 31;
  const int col0 = wave * 16;
  if (col0 >= Nc) return;                  // wave-uniform: EXEC stays all-1s for WMMA

  const int n    = col0 + (lane & 15);     // B/C/D column owned by this lane
  const int m    = lane & 15;              // A row owned by this lane
  const int koff = (lane >> 4) << 1;       // lanes 0-15 -> k+{0,1}; lanes 16-31 -> k+{2,3}

  v8f acc = {};
#pragma unroll
  for (int kb = 0; kb < N_CH; kb += 4) {
    const int k = kb + koff;
    v2f a, b;
    a.x = As[m * LDA + k];
    a.y = As[m * LDA + k + 1];
    b.x = W[k * Nc + n];
    b.y = W[(k + 1) * Nc + n];
    // D = A(16x4) * B(4x16) + C  -> v_wmma_f32_16x16x4_f32
    acc = __builtin_amdgcn_wmma_f32_16x16x4_f32(
        false, a, false, b, (short)0, acc, false, false);
  }

  const float bv = bias ? bias[n] : 0.0f;
  const int mbase = row0 + ((lane >> 4) << 3);   // lanes 16-31 hold rows r+8
#pragma unroll
  for (int r = 0; r < 8; ++r)
    C[(size_t)(mbase + r) * Nc + n] = acc[r] + bv;
}

// ---------------- LayerNorm(128) + ReLU : one wave per node ----------------

__global__ __launch_bounds__(256) void ln_relu(
    const float* __restrict__ h, const float* __restrict__ g,
    const float* __restrict__ b, float* __restrict__ t, int N) {
  int node = blockIdx.x * 8 + (threadIdx.x >> 5);
  int lane = threadIdx.x & 31;
  if (node >= N) return;
  float4 v = ((const float4*)(h + (size_t)node * N_CH))[lane];

  float s = v.x + v.y + v.z + v.w;
#pragma unroll
  for (int off = 16; off > 0; off >>= 1) s += __shfl_xor(s, off, 32);
  float mu = s * (1.0f / 128.0f);

  float dx = v.x - mu, dy = v.y - mu, dz = v.z - mu, dw = v.w - mu;
  float q = dx * dx + dy * dy + dz * dz + dw * dw;
#pragma unroll
  for (int off = 16; off > 0; off >>= 1) q += __shfl_xor(q, off, 32);
  float rstd = rsqrtf(q * (1.0f / 128.0f) + 1e-5f);

  float4 gg = ((const float4*)g)[lane];
  float4 bb = ((const float4*)b)[lane];
  float4 o;
  o.x = fmaxf(dx * rstd * gg.x + bb.x, 0.0f);
  o.y = fmaxf(dy * rstd * gg.y + bb.y, 0.0f);
  o.z = fmaxf(dz * rstd * gg.z + bb.z, 0.0f);
  o.w = fmaxf(dw * rstd * gg.w + bb.w, 0.0f);
  ((float4*)(t + (size_t)node * N_CH))[lane] = o;
}

// ---------------- agg init: agg = xw * dinv^2 + bias (self-loop + conv bias) ----------------

__global__ void init_agg(const float* __restrict__ xw, const float* __restrict__ dinv,
                         const float* __restrict__ bias, float* __restrict__ agg, int N) {
  int i = blockIdx.x * blockDim.x + threadIdx.x;      // over N*32 float4's
  if (i >= N * 32) return;
  int node = i >> 5, c4 = i & 31;
  float d = dinv[node], d2 = d * d;
  float4 v = ((const float4*)xw)[i];
  float4 bb = ((const float4*)bias)[c4];
  float4 o = {v.x * d2 + bb.x, v.y * d2 + bb.y, v.z * d2 + bb.z, v.w * d2 + bb.w};
  ((float4*)agg)[i] = o;
}

// ---------------- edge scatter: one wave per edge, float4 per lane ----------------

__global__ __launch_bounds__(256) void scatter_edges(
    const float* __restrict__ xw, const int* __restrict__ src,
    const int* __restrict__ dst, const float* __restrict__ ew,
    const float* __restrict__ dinv, float* __restrict__ agg, int E) {
  int wid = (blockIdx.x * blockDim.x + threadIdx.x) >> 5;
  int lane = threadIdx.x & 31;
  int nw = (gridDim.x * blockDim.x) >> 5;
  for (int e = wid; e < E; e += nw) {
    int s = src[e], d = dst[e];
    float coef = dinv[s] * ew[e] * dinv[d];       // uniform loads broadcast within wave
    float4 v = ((const float4*)(xw + (size_t)s * N_CH))[lane];
    float* dp = agg + (size_t)d * N_CH + lane * 4;
    unsafeAtomicAdd(dp + 0, v.x * coef);
    unsafeAtomicAdd(dp + 1, v.y * coef);
    unsafeAtomicAdd(dp + 2, v.z * coef);
    unsafeAtomicAdd(dp + 3, v.w * coef);
  }
}

// ---------------- residual combine: h += agg ----------------

__global__ void combine_add(float* __restrict__ h, const float* __restrict__ agg, int N) {
  int i = blockIdx.x * blockDim.x + threadIdx.x;      // over N*32 float4's
  if (i >= N * 32) return;
  float4 a = ((const float4*)h)[i];
  float4 b = ((const float4*)agg)[i];
  float4 o = {a.x + b.x, a.y + b.y, a.z + b.z, a.w + b.w};
  ((float4*)h)[i] = o;
}

// ---------------- driver ----------------

extern "C" void kernel_launch(void* const* d_in, const int* in_sizes, int n_in,
                              void* d_out, int out_size, void* d_ws, size_t ws_size,
                              hipStream_t stream) {
  const float* x     = (const float*)d_in[0];
  const int*   ei    = (const int*)d_in[1];
  const float* ew    = (const float*)d_in[2];
  const float* encW  = (const float*)d_in[3];
  const float* encB  = (const float*)d_in[4];
  const float* convW = (const float*)d_in[5];
  const float* convB = (const float*)d_in[6];
  const float* normG = (const float*)d_in[7];
  const float* normB = (const float*)d_in[8];
  const float* linW  = (const float*)d_in[9];
  const float* linB  = (const float*)d_in[10];
  float* out = (float*)d_out;

  const int M    = in_sizes[0] / N_CH;   // 50000 nodes (multiple of 16)
  const int E    = in_sizes[2];          // 800000 edges
  const int OUTC = in_sizes[10];         // 112

  const int* srcI = ei;
  const int* dstI = ei + E;

  float* h    = (float*)d_ws;
  float* xw   = h  + (size_t)M * N_CH;
  float* T    = xw + (size_t)M * N_CH;   // shared: LN output / agg accumulator
  float* deg  = T  + (size_t)M * N_CH;
  float* dinv = deg + M;

  const int tpb = 256;
  const int gemm_grid = M / 16;
  const int ew_grid = (M * 32 + tpb - 1) / tpb;   // elementwise float4 kernels
  const int ln_grid = (M + 7) / 8;

  // symmetric normalization coefficients (recomputed every call for determinism)
  zero_f32<<<(M + tpb - 1) / tpb, tpb, 0, stream>>>(deg, M);
  deg_acc<<<1024, tpb, 0, stream>>>(dstI, ew, deg, E);
  make_dinv<<<(M + tpb - 1) / tpb, tpb, 0, stream>>>(deg, dinv, M);

  // encoder: h = x @ enc_W + enc_b
  gemm_wmma_f32<<<gemm_grid, 256, 0, stream>>>(x, encW, encB, h, N_CH);

  // layer 0: h = gcn_conv(h, W0, b0)
  gemm_wmma_f32<<<gemm_grid, 256, 0, stream>>>(h, convW, nullptr, xw, N_CH);
  init_agg<<<ew_grid, tpb, 0, stream>>>(xw, dinv, convB, h, M);        // overwrite h
  scatter_edges<<<2048, tpb, 0, stream>>>(xw, srcI, dstI, ew, dinv, h, E);

  // layers 1..3: h = h + gcn_conv(relu(LN(h)), Wi, bi)
  for (int l = 1; l < 4; ++l) {
    ln_relu<<<ln_grid, 256, 0, stream>>>(h, normG + l * N_CH, normB + l * N_CH, T, M);
    gemm_wmma_f32<<<gemm_grid, 256, 0, stream>>>(T, convW + (size_t)l * N_CH * N_CH,
                                                 nullptr, xw, N_CH);
    init_agg<<<ew_grid, tpb, 0, stream>>>(xw, dinv, convB + l * N_CH, T, M);
    scatter_edges<<<2048, tpb, 0, stream>>>(xw, srcI, dstI, ew, dinv, T, E);
    combine_add<<<ew_grid, tpb, 0, stream>>>(h, T, M);
  }

  // final: out = relu(LN(h, g0, b0)) @ lin_W + lin_b
  ln_relu<<<ln_grid, 256, 0, stream>>>(h, normG, normB, T, M);
  gemm_wmma_f32<<<gemm_grid, 256, 0, stream>>>(T, linW, linB, out, OUTC);
}